// Attention_42348377539209
// MI455X (gfx1250) — compile-verified
//
#include <hip/hip_runtime.h>
#include <math.h>

// CDNA5 (gfx1250) paged GQA decode attention, fp32, wave32.
// Flash-decode split-K: grid = B*HKV*SPLIT blocks (4 waves each) producing
// unnormalized (m, l, O) partials in workspace; a second kernel does the
// 32-way log-sum-exp merge per (batch, kv head).
// Matrix core: V_WMMA_F32_16X16X4_F32 for both QK^T and P*V, with explicit
// 8-deep software pipelining of the global A/B fragment loads.

typedef __attribute__((ext_vector_type(2))) float v2f;
typedef __attribute__((ext_vector_type(8))) float v8f;

#define NB    16      // batch
#define NHQ   32      // query heads
#define NHKV  8       // kv heads
#define GQ    4       // query heads per kv head
#define DD    128     // head dim
#define LL    4096    // max kv length
#define NW    4       // waves per workgroup
#define SPLIT 8       // L-splits per (b, kv head)
#define NGW   (NW * SPLIT)   // global waves (partials) per (b, kv head) = 32
#define SCALE 0.08838834764831845f

__device__ __forceinline__ const float* bcast_ptr(unsigned int lo, unsigned int hi32, int src) {
    unsigned int l2 = (unsigned int)__shfl((int)lo, src, 32);
    unsigned int h2 = (unsigned int)__shfl((int)hi32, src, 32);
    return (const float*)(((unsigned long long)h2 << 32) | (unsigned long long)l2);
}

__global__ __launch_bounds__(32 * NW) void paged_gqa_partial(
    const float* __restrict__ q,
    const float* __restrict__ knew,
    const float* __restrict__ vnew,
    const float* __restrict__ kc,
    const float* __restrict__ vc,
    const int*  __restrict__ slot_mapping,
    const int*  __restrict__ kv_indices,
    const int*  __restrict__ kv_len,
    float* __restrict__ wsO,
    float* __restrict__ wsm,
    float* __restrict__ wsl)
{
    const int blk  = blockIdx.x;          // bh * SPLIT + sp
    const int bh   = blk / SPLIT;
    const int sp   = blk % SPLIT;
    const int b    = bh / NHKV;
    const int h    = bh % NHKV;
    const int tid  = threadIdx.x;
    const int wave = tid >> 5;
    const int lane = tid & 31;
    const int g    = lane & 15;   // head column (valid 0..3) / kv row within chunk
    const int hi   = lane >> 4;   // lane half selects K-offset pair (ISA A/B layout)
    const int kvl  = kv_len[b];
    const int gw   = sp * NW + wave;      // global wave id within (b,h): 0..31

    // slot_mapping in registers for virtual scatter resolution
    int smap[16];
#pragma unroll
    for (int j = 0; j < 16; ++j) smap[j] = slot_mapping[j];

    // resolve a gathered slot index to (K row, V row) pointers, honoring the
    // store_kvcache scatter of the fresh tokens (last matching batch wins)
    auto resolve = [&](int idx, const float*& krp, const float*& vrp) {
        krp = kc + ((size_t)idx * NHKV + h) * DD;
        vrp = vc + ((size_t)idx * NHKV + h) * DD;
#pragma unroll
        for (int j = 0; j < 16; ++j) {
            if (idx == smap[j]) {
                krp = knew + ((size_t)j * NHKV + h) * DD;
                vrp = vnew + ((size_t)j * NHKV + h) * DD;
            }
        }
    };

    // ---- Q B-fragments (loop-invariant). B[k][n] = Q[head n][4s+k].
    // Assumed f32 16x16x4 B layout mirrors A: VGPR0 -> K = 2*hi, VGPR1 -> K = 2*hi+1.
    const float qmask = (g < GQ) ? 1.f : 0.f;
    const float* qrow = q + (size_t)(b * NHQ + h * GQ + (g < GQ ? g : 0)) * DD;
    v2f qf[32];
#pragma unroll
    for (int s = 0; s < 32; ++s) {
        v2f t = *(const v2f*)(qrow + 4 * s + 2 * hi);
        qf[s].x = t.x * qmask;
        qf[s].y = t.y * qmask;
    }

    // Online softmax state (head g, replicated in lanes g and g+16)
    float m_st = -3.0e38f;
    float l_st = 0.f;
    v8f Ot[8] = {};   // O tiles: C-layout [M=head 16 (0..3 valid) x N=16 d-cols]

    const int nchunks = (kvl + 15) >> 4;      // kvl >= 2048 -> all 32 global waves active
    const int* kvi = kv_indices + (size_t)b * LL;

    // ---- prologue: resolve this wave's first chunk
    const float* kr = kc;
    const float* vr = vc;
    if (gw < nchunks) {
        const int pos = gw * 16 + g;
        const int idx = (pos < kvl) ? kvi[pos] : kvi[0];
        resolve(idx, kr, vr);
    }

    for (int c = gw; c < nchunks; c += NGW) {
        // ---- issue next chunk's paged-index gather as early as possible
        const int  cn        = c + NGW;
        const bool have_next = cn < nchunks;
        const int  posn      = cn * 16 + g;
        const int  idx_n     = (have_next && posn < kvl) ? kvi[posn] : kvi[0];

        // ---- S^T = K x Q^T : A = K rows (M=pos, K=dim), 32 steps over D=128.
        // Explicit 8-deep ring buffer keeps ~8 global_load_b64 in flight.
        v2f abuf[8];
#pragma unroll
        for (int s = 0; s < 8; ++s)
            abuf[s] = *(const v2f*)(kr + 4 * s + 2 * hi);

        v8f Ct = {};
#pragma unroll
        for (int s = 0; s < 32; ++s) {
            const v2f a = abuf[s & 7];
            if (s < 24)
                abuf[s & 7] = *(const v2f*)(kr + 4 * (s + 8) + 2 * hi);
            Ct = __builtin_amdgcn_wmma_f32_16x16x4_f32(
                    false, a, false, qf[s], (short)0, Ct, false, false);
        }

        // ---- masked scale + chunk max (C layout: vgpr j -> pos = j + 8*hi, lane -> head)
        float p[8];
        float mloc = -3.0e38f;
#pragma unroll
        for (int j = 0; j < 8; ++j) {
            const int pj = c * 16 + j + 8 * hi;
            const float sv = (pj < kvl) ? Ct[j] * SCALE : -3.0e38f;
            p[j] = sv;
            mloc = fmaxf(mloc, sv);
        }
        mloc = fmaxf(mloc, __shfl_xor(mloc, 16, 32));
        const float m_new = fmaxf(m_st, mloc);
        const float alpha = __expf(m_st - m_new);

        float lsum = 0.f;
#pragma unroll
        for (int j = 0; j < 8; ++j) {
            const int pj = c * 16 + j + 8 * hi;
            const float e = (pj < kvl) ? __expf(p[j] - m_new) : 0.f;
            p[j] = e;
            lsum += e;
        }
        lsum += __shfl_xor(lsum, 16, 32);
        l_st = l_st * alpha + lsum;
        m_st = m_new;

        // rescale valid O rows (heads 0..3 live in C vgprs 0..3, lanes 0..15)
#pragma unroll
        for (int j = 0; j < GQ; ++j) {
            const float aj = __shfl(alpha, j, 32);
#pragma unroll
            for (int t = 0; t < 8; ++t) Ot[t][j] *= aj;
        }

        // ---- P*V : A = P (M=head, K=4 positions). Relayout S^T -> A via one half-swap.
        float dp[8];
#pragma unroll
        for (int j = 0; j < 8; ++j) dp[j] = __shfl_xor(p[j], 16, 32);

        v2f apf[4];
#pragma unroll
        for (int s = 0; s < 4; ++s) {
            const int base = (s & 1) * 4;
            if (s < 2) {  // lanes<16: p[base,base+1]; lanes>=16: dp[base+2,base+3]
                apf[s].x = hi ? dp[base + 2] : p[base];
                apf[s].y = hi ? dp[base + 3] : p[base + 1];
            } else {      // lanes<16: dp[base,base+1]; lanes>=16: p[base+2,base+3]
                apf[s].x = hi ? p[base + 2] : dp[base];
                apf[s].y = hi ? p[base + 3] : dp[base + 1];
            }
        }

        // broadcast all 8 V row pointers (rows 4s+2*hi and 4s+2*hi+1, s=0..3)
        const unsigned long long vru = (unsigned long long)vr;
        const unsigned int vlo  = (unsigned int)vru;
        const unsigned int vhiw = (unsigned int)(vru >> 32);
        const float* vrow[8];
#pragma unroll
        for (int s = 0; s < 4; ++s) {
            const int r0 = 4 * s + 2 * hi;
            vrow[2 * s]     = bcast_ptr(vlo, vhiw, r0);
            vrow[2 * s + 1] = bcast_ptr(vlo, vhiw, r0 + 1);
        }

        // 8-deep ring buffer of V B-fragments (2 b32 loads per slot)
        v2f vbuf[8];
#pragma unroll
        for (int i = 0; i < 8; ++i) {
            vbuf[i].x = vrow[0][i * 16 + g];
            vbuf[i].y = vrow[1][i * 16 + g];
        }
#pragma unroll
        for (int i = 0; i < 32; ++i) {    // i = s*8 + t
            const int s = i >> 3;
            const int t = i & 7;
            const v2f bf = vbuf[t];
            if (i < 24) {
                const int ii = i + 8;
                const int sn = ii >> 3;
                vbuf[t].x = vrow[2 * sn][(ii & 7) * 16 + g];
                vbuf[t].y = vrow[2 * sn + 1][(ii & 7) * 16 + g];
            }
            Ot[t] = __builtin_amdgcn_wmma_f32_16x16x4_f32(
                        false, apf[s], false, bf, (short)0, Ot[t], false, false);
        }

        // ---- retire the pipelined index: resolve + prefetch next chunk's rows
        if (have_next) {
            resolve(idx_n, kr, vr);
            __builtin_prefetch(kr, 0, 3);        // global_prefetch_b8: warm next K row
            __builtin_prefetch(kr + 64, 0, 3);
            __builtin_prefetch(vr, 0, 3);        // and next V row
            __builtin_prefetch(vr + 64, 0, 3);
        }
    }

    // ---- publish this wave's unnormalized partial (no LDS, no barrier)
    const int pw = bh * NGW + gw;
    if (lane < 16) {
#pragma unroll
        for (int t = 0; t < 8; ++t)
#pragma unroll
            for (int j = 0; j < GQ; ++j)
                wsO[((size_t)pw * GQ + j) * DD + t * 16 + lane] = Ot[t][j];
    }
    if (lane < GQ) {
        wsm[pw * GQ + lane] = m_st;
        wsl[pw * GQ + lane] = l_st;
    }
}

// ---- second pass: 32-way log-sum-exp merge per (b, kv head) -----------------
__global__ __launch_bounds__(128) void paged_gqa_merge(
    const float* __restrict__ wsO,
    const float* __restrict__ wsm,
    const float* __restrict__ wsl,
    float* __restrict__ out)
{
    const int bh  = blockIdx.x;
    const int b   = bh / NHKV;
    const int h   = bh % NHKV;
    const int tid = threadIdx.x;

#pragma unroll
    for (int e = 0; e < 4; ++e) {
        const int io = tid * 4 + e;       // 0..511
        const int gg = io >> 7;           // head within group
        const int d  = io & 127;

        float M = -3.0e38f;
#pragma unroll
        for (int w = 0; w < NGW; ++w)
            M = fmaxf(M, wsm[(bh * NGW + w) * GQ + gg]);

        float Lsum = 0.f, acc = 0.f;
#pragma unroll
        for (int w = 0; w < NGW; ++w) {
            const int pw = bh * NGW + w;
            const float aw = __expf(wsm[pw * GQ + gg] - M);
            Lsum += aw * wsl[pw * GQ + gg];
            acc  += aw * wsO[((size_t)pw * GQ + gg) * DD + d];
        }
        out[(size_t)b * NHQ * DD + (size_t)(h * GQ + gg) * DD + d] = acc / Lsum;
    }
}

extern "C" void kernel_launch(void* const* d_in, const int* in_sizes, int n_in,
                              void* d_out, int out_size, void* d_ws, size_t ws_size,
                              hipStream_t stream) {
    const float* q    = (const float*)d_in[0];
    const float* k    = (const float*)d_in[1];
    const float* v    = (const float*)d_in[2];
    const float* kc   = (const float*)d_in[3];
    const float* vc   = (const float*)d_in[4];
    const int*   slot = (const int*)d_in[5];
    const int*   kvi  = (const int*)d_in[6];
    const int*   kvl  = (const int*)d_in[7];
    float* outp = (float*)d_out;

    // workspace carve-up: 4096 partials x (4 heads x 128 dims + m + l)  (~8 MB)
    const int nPart = NB * NHKV * NGW;               // 4096
    float* wsO = (float*)d_ws;                       // nPart * GQ * DD floats
    float* wsm = wsO + (size_t)nPart * GQ * DD;      // nPart * GQ floats
    float* wsl = wsm + (size_t)nPart * GQ;           // nPart * GQ floats

    paged_gqa_partial<<<dim3(NB * NHKV * SPLIT), dim3(32 * NW), 0, stream>>>(
        q, k, v, kc, vc, slot, kvi, kvl, wsO, wsm, wsl);
    paged_gqa_merge<<<dim3(NB * NHKV), dim3(128), 0, stream>>>(
        wsO, wsm, wsl, outp);
}